// Qkernel_47940424958528
// MI455X (gfx1250) — compile-verified
//
#include <hip/hip_runtime.h>

typedef __attribute__((ext_vector_type(2))) float v2f;
typedef __attribute__((ext_vector_type(8))) float v8f;

#define NQ      18
#define LAYERS  6
#define NSTATES (1 << NQ)   // 262144 amplitudes per state

__device__ __forceinline__ float2 cmul(float2 a, float2 b) {
    return make_float2(a.x * b.x - a.y * b.y, a.x * b.y + a.y * b.x);
}

// ---------------------------------------------------------------------------
// Build fused per-(state,layer,qubit) gates: G = RZ(tz) * RY(ty) * RX(tx)
// gates layout: [2][LAYERS][NQ][4] float2  (g00,g01,g10,g11)
// ---------------------------------------------------------------------------
__global__ void prep_gates_kernel(const float* __restrict__ x1,
                                  const float* __restrict__ x2,
                                  const float* __restrict__ iscal,
                                  const float* __restrict__ var,
                                  float2* __restrict__ gates) {
    int t = blockIdx.x * blockDim.x + threadIdx.x;
    if (t >= 2 * LAYERS * NQ) return;
    int q = t % NQ;
    int l = (t / NQ) % LAYERS;
    int s = t / (NQ * LAYERS);
    const float* x = s ? x2 : x1;
    float tx = iscal[l * NQ + q] * x[q];
    float ty = var[l * 2 * NQ + q];
    float tz = var[l * 2 * NQ + NQ + q];
    float cx = cosf(0.5f * tx), sx = sinf(0.5f * tx);
    float cy = cosf(0.5f * ty), sy = sinf(0.5f * ty);
    float cz = cosf(0.5f * tz), sz = sinf(0.5f * tz);
    // U = RY * RX   (RX = [[c,-i s],[-i s,c]], RY = [[c,-s],[s,c]])
    float2 u00 = make_float2( cy * cx,  sy * sx);
    float2 u01 = make_float2(-sy * cx, -cy * sx);
    float2 u10 = make_float2( sy * cx, -cy * sx);
    float2 u11 = make_float2( cy * cx, -sy * sx);
    float2 em  = make_float2(cz, -sz);   // e^{-i tz/2}
    float2 ep  = make_float2(cz,  sz);   // e^{+i tz/2}
    float2* g = gates + t * 4;
    g[0] = cmul(em, u00);
    g[1] = cmul(em, u01);
    g[2] = cmul(ep, u10);
    g[3] = cmul(ep, u11);
}

// |0..0> for both states
__global__ void init_kernel(float2* __restrict__ psi) {
    int i = blockIdx.x * blockDim.x + threadIdx.x;   // 0 .. 2*NSTATES-1
    float v = ((i & (NSTATES - 1)) == 0) ? 1.f : 0.f;
    psi[i] = make_float2(v, 0.f);
}

// ---------------------------------------------------------------------------
// One fused 2x2-complex gate (4x4 real matrix M) applied to 16 amplitude
// pairs per wave via V_WMMA_F32_16X16X4_F32.
//   A (16x4): rows 0-3 = M, rows 4-15 = 0.
//     lane L<16 holds A[L][0],A[L][1]; lane L>=16 holds A[L-16][2],A[L-16][3]
//   B (4x16): col n = (a0r,a0i,a1r,a1i) of pair n.
//     lane n<16 supplies K0,K1 (= amp0.x, amp0.y);
//     lane 16+n supplies K2,K3 (= amp1.x, amp1.y)
//   D: rows 0-3 (VGPR0-3, lanes 0-15) = (b0r,b0i,b1r,b1i)
// M = [[g00r,-g00i,g01r,-g01i],
//      [g00i, g00r,g01i, g01r],
//      [g10r,-g10i,g11r,-g11i],
//      [g10i, g10r,g11i, g11r]]
// ---------------------------------------------------------------------------
__device__ __forceinline__ v8f gate_wmma(float2 g00, float2 g01,
                                         float2 g10, float2 g11,
                                         int lane, float2 amp) {
    int r  = lane & 15;
    int hi = lane >> 4;                      // 0 -> cols {0,1}, 1 -> cols {2,3}
    float2 gs = (r & 2) ? (hi ? g11 : g10) : (hi ? g01 : g00);
    float ax = (r & 1) ? gs.y :  gs.x;
    float ay = (r & 1) ? gs.x : -gs.y;
    if (r >= 4) { ax = 0.f; ay = 0.f; }
    v2f a; a.x = ax;    a.y = ay;
    v2f b; b.x = amp.x; b.y = amp.y;
    v8f c = {0.f, 0.f, 0.f, 0.f, 0.f, 0.f, 0.f, 0.f};
    return __builtin_amdgcn_wmma_f32_16x16x4_f32(
        /*neg_a=*/false, a, /*neg_b=*/false, b,
        /*c_mod=*/(short)0, c, /*reuse_a=*/false, /*reuse_b=*/false);
}

// ---------------------------------------------------------------------------
// Qubits 9..17 (in-block strides 256..1). 512 contiguous amps per workgroup,
// staged in LDS; 9 fused gates with barriers between. Grid: 2 states * 512.
// ---------------------------------------------------------------------------
__global__ void __launch_bounds__(256)
apply_low_kernel(float2* __restrict__ psi, const float2* __restrict__ gates,
                 int layer) {
    __shared__ float2 lds[512];
    int tid = threadIdx.x;
    int s   = blockIdx.x >> 9;
    int blk = blockIdx.x & 511;
    float2* base = psi + s * NSTATES + blk * 512;
    lds[tid]       = base[tid];
    lds[tid + 256] = base[tid + 256];
    __syncthreads();
    int lane = tid & 31;
    int w    = tid >> 5;
    int col  = lane & 15;
    const float2* gl = gates + (size_t)(s * LAYERS + layer) * NQ * 4;
    for (int q = 9; q < NQ; ++q) {
        int k = 17 - q;                 // stride = 2^k, k in 8..0
        int d = 1 << k;
        float2 g00 = gl[q * 4 + 0], g01 = gl[q * 4 + 1];
        float2 g10 = gl[q * 4 + 2], g11 = gl[q * 4 + 3];
        for (int rr = 0; rr < 2; ++rr) {         // 8 waves * 16 pairs * 2
            int p    = rr * 128 + w * 16 + col;  // pair id 0..255
            int lo   = p & (d - 1);
            int i0   = ((p >> k) << (k + 1)) | lo;
            int i1   = i0 + d;
            float2 amp = lds[(lane < 16) ? i0 : i1];
            v8f dd = gate_wmma(g00, g01, g10, g11, lane, amp);
            if (lane < 16) {
                lds[i0] = make_float2(dd[0], dd[1]);
                lds[i1] = make_float2(dd[2], dd[3]);
            }
        }
        __syncthreads();
    }
    base[tid]       = lds[tid];
    base[tid + 256] = lds[tid + 256];
}

// ---------------------------------------------------------------------------
// Qubits 0..8 (amp strides 2^17..2^9). Workgroup owns 8 low-index values c
// and all 512 high-index values h: amp(h,c) at h*512 + c0 + cc. 32 KB LDS.
// CZ-chain diagonal sign folded into write-back. Grid: 2 states * 64.
// ---------------------------------------------------------------------------
__global__ void __launch_bounds__(256)
apply_high_kernel(float2* __restrict__ psi, const float2* __restrict__ gates,
                  int layer) {
    __shared__ float2 lds[4096];                 // [h][cc], linear = h*8+cc
    int tid = threadIdx.x;
    int s   = blockIdx.x >> 6;
    int c0  = (blockIdx.x & 63) * 8;
    float2* base = psi + s * NSTATES;
    for (int j = 0; j < 16; ++j) {
        int lin = j * 256 + tid;
        int h = lin >> 3, cc = lin & 7;
        lds[lin] = base[h * 512 + c0 + cc];
    }
    __syncthreads();
    int lane = tid & 31;
    int w    = tid >> 5;
    int col  = lane & 15;
    const float2* gl = gates + (size_t)(s * LAYERS + layer) * NQ * 4;
    for (int q = 0; q <= 8; ++q) {
        int k = 8 - q;                  // h-stride = 2^k, k in 8..0
        int d = 1 << k;
        float2 g00 = gl[q * 4 + 0], g01 = gl[q * 4 + 1];
        float2 g10 = gl[q * 4 + 2], g11 = gl[q * 4 + 3];
        for (int rr = 0; rr < 16; ++rr) {        // 2048 pairs / 128 per round
            int p  = rr * 128 + w * 16 + col;    // 0..2047
            int ph = p >> 3, cc = p & 7;
            int lo = ph & (d - 1);
            int h0 = ((ph >> k) << (k + 1)) | lo;
            int h1 = h0 + d;
            int i0 = h0 * 8 + cc, i1 = h1 * 8 + cc;
            float2 amp = lds[(lane < 16) ? i0 : i1];
            v8f dd = gate_wmma(g00, g01, g10, g11, lane, amp);
            if (lane < 16) {
                lds[i0] = make_float2(dd[0], dd[1]);
                lds[i1] = make_float2(dd[2], dd[3]);
            }
        }
        __syncthreads();
    }
    // CZ chain: sign = (-1)^popcount(idx & (idx>>1)) over the 18-bit index
    for (int j = 0; j < 16; ++j) {
        int lin = j * 256 + tid;
        int h = lin >> 3, cc = lin & 7;
        int gidx = h * 512 + c0 + cc;
        unsigned tb = (unsigned)gidx & ((unsigned)gidx >> 1);
        float sgn = (__popc(tb) & 1) ? -1.f : 1.f;
        float2 a = lds[lin];
        base[gidx] = make_float2(sgn * a.x, sgn * a.y);
    }
}

// ---------------------------------------------------------------------------
// ov = sum conj(psi2[i]) * psi1[i];  out = |ov|^2   (deterministic 2-stage)
// ---------------------------------------------------------------------------
__global__ void __launch_bounds__(256)
reduce1_kernel(const float2* __restrict__ psi, float2* __restrict__ partials) {
    __shared__ float2 sm[256];
    int tid = threadIdx.x;
    const float2* p1 = psi;
    const float2* p2 = psi + NSTATES;
    float re = 0.f, im = 0.f;
    int base = blockIdx.x * 1024 + tid;
    for (int j = 0; j < 4; ++j) {
        int i = base + j * 256;
        float2 a1 = p1[i], a2 = p2[i];
        re += a2.x * a1.x + a2.y * a1.y;
        im += a2.x * a1.y - a2.y * a1.x;
    }
    sm[tid] = make_float2(re, im);
    __syncthreads();
    for (int s2 = 128; s2 > 0; s2 >>= 1) {
        if (tid < s2) {
            sm[tid].x += sm[tid + s2].x;
            sm[tid].y += sm[tid + s2].y;
        }
        __syncthreads();
    }
    if (tid == 0) partials[blockIdx.x] = sm[0];
}

__global__ void __launch_bounds__(256)
reduce2_kernel(const float2* __restrict__ partials, float* __restrict__ out) {
    __shared__ float2 sm[256];
    int tid = threadIdx.x;
    sm[tid] = partials[tid];
    __syncthreads();
    for (int s2 = 128; s2 > 0; s2 >>= 1) {
        if (tid < s2) {
            sm[tid].x += sm[tid + s2].x;
            sm[tid].y += sm[tid + s2].y;
        }
        __syncthreads();
    }
    if (tid == 0) out[0] = sm[0].x * sm[0].x + sm[0].y * sm[0].y;
}

// ---------------------------------------------------------------------------
extern "C" void kernel_launch(void* const* d_in, const int* in_sizes, int n_in,
                              void* d_out, int out_size, void* d_ws, size_t ws_size,
                              hipStream_t stream) {
    (void)in_sizes; (void)n_in; (void)out_size; (void)ws_size;
    const float* x1    = (const float*)d_in[0];
    const float* x2    = (const float*)d_in[1];
    const float* iscal = (const float*)d_in[2];
    const float* var   = (const float*)d_in[3];

    float2* psi      = (float2*)d_ws;                    // 2 * NSTATES
    float2* gates    = psi + 2 * NSTATES;                // 2*6*18*4
    float2* partials = gates + 2 * LAYERS * NQ * 4;      // 256

    prep_gates_kernel<<<1, 256, 0, stream>>>(x1, x2, iscal, var, gates);
    init_kernel<<<(2 * NSTATES) / 256, 256, 0, stream>>>(psi);
    for (int l = 0; l < LAYERS; ++l) {
        apply_low_kernel <<<1024, 256, 0, stream>>>(psi, gates, l);
        apply_high_kernel<<<128,  256, 0, stream>>>(psi, gates, l);
    }
    reduce1_kernel<<<256, 256, 0, stream>>>(psi, partials);
    reduce2_kernel<<<1,   256, 0, stream>>>(partials, (float*)d_out);
}